// RolandLayer_64218351010254
// MI455X (gfx1250) — compile-verified
//
#include <hip/hip_runtime.h>
#include <math.h>

typedef float v2f __attribute__((ext_vector_type(2)));
typedef float v8f __attribute__((ext_vector_type(8)));

#define D 128
#define LDA 132  // padded LDS row stride for A tiles (floats): conflict-free, 16B align
#define TS 34    // padded stride for transposed weight chunk: even (8B align), conflict-free b64

__device__ __forceinline__ float sig_(float v) { return 1.0f / (1.0f + __expf(-v)); }

// ---------------------------------------------------------------- init
__global__ void k_init(float* __restrict__ agg, float* __restrict__ deg,
                       float* __restrict__ stats, int N) {
  const long long nd = (long long)N * D;
  const long long total = nd + N + 2 * D;
  for (long long i = (long long)blockIdx.x * blockDim.x + threadIdx.x; i < total;
       i += (long long)gridDim.x * blockDim.x) {
    if (i < nd) agg[i] = 0.0f;
    else if (i < nd + N) deg[i - nd] = 1.0f;       // self-loop contributes 1
    else stats[i - nd - N] = 0.0f;
  }
}

// ---------------------------------------------------------------- degree
__global__ void k_deg(const int* __restrict__ ei, float* __restrict__ deg, int E) {
  int e = blockIdx.x * blockDim.x + threadIdx.x;
  if (e < E) unsafeAtomicAdd(&deg[ei[E + e]], 1.0f);   // cols = targets
}

__global__ void k_dinv(float* __restrict__ deg, int N) {
  int i = blockIdx.x * blockDim.x + threadIdx.x;
  if (i < N) deg[i] = rsqrtf(deg[i]);
}

// ---------------------------------------------------------------- xt = x @ W_gcn
__launch_bounds__(256, 1)
__global__ void k_gcn_gemm(const float* __restrict__ x, const float* __restrict__ Wg,
                           float* __restrict__ xt, int N) {
  __shared__ __align__(16) float sA[D][LDA];
  __shared__ __align__(16) float sWt[D][TS];   // transposed 32-K chunk of W
  const int tid = threadIdx.x;
  const int row0 = blockIdx.x * D;  // 128 rows per block

  for (int i = tid; i < D * 32; i += 256) {   // 32 float4 per row
    int r = i >> 5, c4 = (i & 31) << 2;
    int gr = row0 + r;
    float4 v = make_float4(0.f, 0.f, 0.f, 0.f);
    if (gr < N) v = *(const float4*)(x + (size_t)gr * D + c4);
    sA[r][c4] = v.x; sA[r][c4 + 1] = v.y; sA[r][c4 + 2] = v.z; sA[r][c4 + 3] = v.w;
  }

  const int lane = tid & 31, wave = tid >> 5;
  const int rA = (wave << 4) + (lane & 15);
  const int koff = (lane >> 4) << 1;        // K phase for A/B operands
  const int nlan = lane & 15;
  const int rstore = (wave << 4) + ((lane >> 4) << 3);

  v8f acc[8];
#pragma unroll
  for (int nt = 0; nt < 8; ++nt) acc[nt] = (v8f){};

  for (int kc = 0; kc < 4; ++kc) {          // K = 128 in chunks of 32
    __syncthreads();
    for (int i = tid; i < 32 * 32; i += 256) {        // 32 k-rows x 32 float4
      int r = i >> 5, c4 = (i & 31) << 2;
      float4 w4 = *(const float4*)(Wg + (size_t)((kc << 5) + r) * D + c4);
      sWt[c4][r] = w4.x; sWt[c4 + 1][r] = w4.y; sWt[c4 + 2][r] = w4.z; sWt[c4 + 3][r] = w4.w;
    }
    if (kc + 1 < 4)
      __builtin_prefetch((const char*)(Wg + (size_t)((kc + 1) << 5) * D) + tid * 128, 0, 3);
    __syncthreads();
    const int kg0 = kc << 5;
#pragma unroll
    for (int nt = 0; nt < 8; ++nt) {
      const int col = (nt << 4) + nlan;
#pragma unroll
      for (int k4 = 0; k4 < 8; ++k4) {
        const int kw = (k4 << 2) + koff;
        v2f a = *(const v2f*)&sA[rA][kg0 + kw];
        v2f b = *(const v2f*)&sWt[col][kw];
        acc[nt] = __builtin_amdgcn_wmma_f32_16x16x4_f32(false, a, false, b, (short)0,
                                                        acc[nt], false, false);
      }
    }
  }

#pragma unroll
  for (int nt = 0; nt < 8; ++nt) {
    const int col = (nt << 4) + nlan;
#pragma unroll
    for (int i = 0; i < 8; ++i) {
      int gr = row0 + rstore + i;
      if (gr < N) xt[(size_t)gr * D + col] = acc[nt][i];
    }
  }
}

// ---------------------------------------------------------------- edge scatter-add
__global__ void k_scatter(const int* __restrict__ ei, const float* __restrict__ xt,
                          const float* __restrict__ dinv, float* __restrict__ agg,
                          int N, int E) {
  const int wid = (int)(((long long)blockIdx.x * blockDim.x + threadIdx.x) >> 5);
  const int lane = threadIdx.x & 31;
  const int tot = E + N;
  if (wid >= tot) return;
  int u, v;
  if (wid < E) { u = ei[wid]; v = ei[E + wid]; }
  else { u = v = wid - E; }                      // self loop
  const float w = dinv[u] * dinv[v];
  const float4 val = *(const float4*)(xt + (size_t)u * D + (lane << 2));
  float* dst = agg + (size_t)v * D + (lane << 2);
  unsafeAtomicAdd(dst + 0, val.x * w);
  unsafeAtomicAdd(dst + 1, val.y * w);
  unsafeAtomicAdd(dst + 2, val.z * w);
  unsafeAtomicAdd(dst + 3, val.w * w);
}

// ---------------------------------------------------------------- batchnorm stats
__global__ void k_bnstats(const float* __restrict__ agg, const float* __restrict__ bg,
                          float* __restrict__ stats, int N) {
  const int c = threadIdx.x;                 // blockDim = 128
  const int r0 = blockIdx.x * 256;
  const int rend = min(r0 + 256, N);
  const float b = bg[c];
  float s = 0.f, q = 0.f;
  for (int r = r0; r < rend; ++r) {
    float v = agg[(size_t)r * D + c] + b;
    s += v; q += v * v;
  }
  unsafeAtomicAdd(&stats[c], s);
  unsafeAtomicAdd(&stats[D + c], q);
}

// ---------------------------------------------------------------- fused BN+PReLU+GRU
__launch_bounds__(256, 1)
__global__ void k_gru(const float* __restrict__ agg, const float* __restrict__ Hm,
                      const float* __restrict__ bg, const float* __restrict__ gamma,
                      const float* __restrict__ beta, const float* __restrict__ pa,
                      const float* __restrict__ Wz, const float* __restrict__ bz,
                      const float* __restrict__ Wr, const float* __restrict__ br,
                      const float* __restrict__ Wh, const float* __restrict__ bh,
                      const float* __restrict__ stats, float* __restrict__ out, int N) {
  __shared__ __align__(16) float sH[D][LDA];   // h after BN+PReLU
  __shared__ __align__(16) float sG[D][LDA];   // H, later overwritten with sig(R)*H
  __shared__ __align__(16) float sWt[D][TS];   // transposed 32-K weight chunk
  __shared__ float sScale[D], sShift[D];

  const int tid = threadIdx.x;
  if (tid < D) {
    float invN = 1.0f / (float)N;
    float mean = stats[tid] * invN;
    float var  = stats[D + tid] * invN - mean * mean;
    float inv  = rsqrtf(var + 1e-5f);
    float sc   = gamma[tid] * inv;
    sScale[tid] = sc;
    sShift[tid] = beta[tid] + (bg[tid] - mean) * sc;   // (agg+b-mean)*sc+beta
  }
  __syncthreads();

  const float alpha = pa[0];
  const int row0 = blockIdx.x * D;
  for (int i = tid; i < D * 32; i += 256) {
    int r = i >> 5, c4 = (i & 31) << 2;
    int gr = row0 + r;
    float4 v = make_float4(0.f, 0.f, 0.f, 0.f);
    float4 h4 = make_float4(0.f, 0.f, 0.f, 0.f);
    if (gr < N) {
      v = *(const float4*)(agg + (size_t)gr * D + c4);
      h4 = *(const float4*)(Hm + (size_t)gr * D + c4);
    }
    float t[4] = {v.x, v.y, v.z, v.w};
#pragma unroll
    for (int j = 0; j < 4; ++j) {
      float u = t[j] * sScale[c4 + j] + sShift[c4 + j];
      sH[r][c4 + j] = u > 0.f ? u : alpha * u;
    }
    sG[r][c4] = h4.x; sG[r][c4 + 1] = h4.y; sG[r][c4 + 2] = h4.z; sG[r][c4 + 3] = h4.w;
  }
  // first __syncthreads() inside the GEMM chunk loop covers staging

  const int lane = tid & 31, wave = tid >> 5;
  const int rA = (wave << 4) + (lane & 15);
  const int koff = (lane >> 4) << 1;
  const int nlan = lane & 15;
  const int rstore = (wave << 4) + ((lane >> 4) << 3);

  v8f acc[8];

  // ---- Z GEMM then R GEMM (shared accumulator set) ----
  for (int m = 0; m < 2; ++m) {
    const float* Wm = m ? Wr : Wz;
    const float* bm = m ? br : bz;
#pragma unroll
    for (int nt = 0; nt < 8; ++nt) {
      float b = bm[(nt << 4) + nlan];
      acc[nt] = (v8f){b, b, b, b, b, b, b, b};
    }
    for (int kc = 0; kc < 8; ++kc) {          // K = 256 in chunks of 32
      __syncthreads();
      for (int i = tid; i < 32 * 32; i += 256) {
        int r = i >> 5, c4 = (i & 31) << 2;
        float4 w4 = *(const float4*)(Wm + (size_t)((kc << 5) + r) * D + c4);
        sWt[c4][r] = w4.x; sWt[c4 + 1][r] = w4.y; sWt[c4 + 2][r] = w4.z; sWt[c4 + 3][r] = w4.w;
      }
      if (kc + 1 < 8)
        __builtin_prefetch((const char*)(Wm + (size_t)((kc + 1) << 5) * D) + tid * 128, 0, 3);
      __syncthreads();
      const float(*A)[LDA] = (kc < 4) ? sH : sG;
      const int kl0 = (kc & 3) << 5;
#pragma unroll
      for (int nt = 0; nt < 8; ++nt) {
        const int col = (nt << 4) + nlan;
#pragma unroll
        for (int k4 = 0; k4 < 8; ++k4) {
          const int kw = (k4 << 2) + koff;
          v2f a = *(const v2f*)&A[rA][kl0 + kw];
          v2f b = *(const v2f*)&sWt[col][kw];
          acc[nt] = __builtin_amdgcn_wmma_f32_16x16x4_f32(false, a, false, b, (short)0,
                                                          acc[nt], false, false);
        }
      }
    }
    if (m == 0) {  // park sigmoid(Z) in d_out
#pragma unroll
      for (int nt = 0; nt < 8; ++nt) {
        const int col = (nt << 4) + nlan;
#pragma unroll
        for (int i = 0; i < 8; ++i) {
          int gr = row0 + rstore + i;
          if (gr < N) out[(size_t)gr * D + col] = sig_(acc[nt][i]);
        }
      }
    } else {       // sG <- sigmoid(R) * H  (wave-local rows)
#pragma unroll
      for (int nt = 0; nt < 8; ++nt) {
        const int col = (nt << 4) + nlan;
#pragma unroll
        for (int i = 0; i < 8; ++i) {
          int rl = rstore + i;
          sG[rl][col] = sig_(acc[nt][i]) * sG[rl][col];
        }
      }
    }
  }

  // ---- H_tilde GEMM: [h | sig(R)*H] @ Wh + bh ----
#pragma unroll
  for (int nt = 0; nt < 8; ++nt) {
    float b = bh[(nt << 4) + nlan];
    acc[nt] = (v8f){b, b, b, b, b, b, b, b};
  }
  for (int kc = 0; kc < 8; ++kc) {
    __syncthreads();
    for (int i = tid; i < 32 * 32; i += 256) {
      int r = i >> 5, c4 = (i & 31) << 2;
      float4 w4 = *(const float4*)(Wh + (size_t)((kc << 5) + r) * D + c4);
      sWt[c4][r] = w4.x; sWt[c4 + 1][r] = w4.y; sWt[c4 + 2][r] = w4.z; sWt[c4 + 3][r] = w4.w;
    }
    if (kc + 1 < 8)
      __builtin_prefetch((const char*)(Wh + (size_t)((kc + 1) << 5) * D) + tid * 128, 0, 3);
    __syncthreads();
    const float(*A)[LDA] = (kc < 4) ? sH : sG;
    const int kl0 = (kc & 3) << 5;
#pragma unroll
    for (int nt = 0; nt < 8; ++nt) {
      const int col = (nt << 4) + nlan;
#pragma unroll
      for (int k4 = 0; k4 < 8; ++k4) {
        const int kw = (k4 << 2) + koff;
        v2f a = *(const v2f*)&A[rA][kl0 + kw];
        v2f b = *(const v2f*)&sWt[col][kw];
        acc[nt] = __builtin_amdgcn_wmma_f32_16x16x4_f32(false, a, false, b, (short)0,
                                                        acc[nt], false, false);
      }
    }
  }

  // ---- epilogue: out = z*H + (1-z)*tanh(htilde) ----
#pragma unroll
  for (int nt = 0; nt < 8; ++nt) {
    const int col = (nt << 4) + nlan;
#pragma unroll
    for (int i = 0; i < 8; ++i) {
      int gr = row0 + rstore + i;
      if (gr < N) {
        float z = out[(size_t)gr * D + col];
        float hp = Hm[(size_t)gr * D + col];
        float ht = tanhf(acc[nt][i]);
        out[(size_t)gr * D + col] = z * hp + (1.0f - z) * ht;
      }
    }
  }
}

// ---------------------------------------------------------------- launcher
extern "C" void kernel_launch(void* const* d_in, const int* in_sizes, int n_in,
                              void* d_out, int out_size, void* d_ws, size_t ws_size,
                              hipStream_t stream) {
  const float* x     = (const float*)d_in[0];
  const int*   ei    = (const int*)d_in[1];     // JAX default: int32 on device
  const float* Hm    = (const float*)d_in[2];
  const float* Wg    = (const float*)d_in[3];
  const float* bg    = (const float*)d_in[4];
  const float* gamma = (const float*)d_in[5];
  const float* beta  = (const float*)d_in[6];
  const float* pa    = (const float*)d_in[7];
  const float* Wz    = (const float*)d_in[8];
  const float* bz    = (const float*)d_in[9];
  const float* Wr    = (const float*)d_in[10];
  const float* br    = (const float*)d_in[11];
  const float* Wh    = (const float*)d_in[12];
  const float* bh    = (const float*)d_in[13];
  float* out = (float*)d_out;

  const int N = in_sizes[0] / D;
  const int E = in_sizes[1] / 2;

  float* xt    = (float*)d_ws;
  float* agg   = xt + (size_t)N * D;
  float* deg   = agg + (size_t)N * D;     // becomes dinv
  float* stats = deg + N;                 // 2*D floats (sum, sumsq)

  k_init<<<1024, 256, 0, stream>>>(agg, deg, stats, N);
  k_deg<<<(E + 255) / 256, 256, 0, stream>>>(ei, deg, E);
  k_dinv<<<(N + 255) / 256, 256, 0, stream>>>(deg, N);

  const int rb = (N + D - 1) / D;
  k_gcn_gemm<<<rb, 256, 0, stream>>>(x, Wg, xt, N);

  const int waves = E + N;
  k_scatter<<<(waves + 7) / 8, 256, 0, stream>>>(ei, xt, deg, agg, N, E);

  k_bnstats<<<(N + 255) / 256, 128, 0, stream>>>(agg, bg, stats, N);

  k_gru<<<rb, 256, 0, stream>>>(agg, Hm, bg, gamma, beta, pa, Wz, bz, Wr, br, Wh, bh,
                                stats, out, N);
}